// LayerStacks_22127671509134
// MI455X (gfx1250) — compile-verified
//
#include <hip/hip_runtime.h>

typedef __attribute__((ext_vector_type(2))) float v2f;
typedef __attribute__((ext_vector_type(8))) float v8f;

#define NB 60                       // buckets (COUNT)
#define W1P_STRIDE 260              // W1 K padded 257 -> 260 (even, mult of 4)
#define W1P_FLOATS (NB * 16 * W1P_STRIDE)
// ws layout in ints
#define WS_COUNTS 0                 // 60
#define WS_TSTART 64                // 61 (tile start per bucket, tile units)
#define WS_CURSOR 128               // 60 (scatter cursors, sample-slot units)
#define WS_W1P    192               // repacked W1: [60][16][260] floats
#define WS_PERM   (WS_W1P + W1P_FLOATS)  // maxTiles*16 sample ids, -1 = pad

static __device__ __forceinline__ v8f wmma4(v2f a, v2f b, v8f c) {
  // D = A(16x4,f32) * B(4x16,f32) + C(16x16,f32)
  return __builtin_amdgcn_wmma_f32_16x16x4_f32(false, a, false, b, (short)0, c,
                                               false, false);
}

__global__ void k_init(int* ws, int permCap) {
  int i = blockIdx.x * blockDim.x + threadIdx.x;
  if (i < NB) ws[WS_COUNTS + i] = 0;
  int stride = gridDim.x * blockDim.x;
  for (int p = i; p < permCap; p += stride) ws[WS_PERM + p] = -1;
}

__global__ void k_repack(const float* __restrict__ W1, int* ws) {
  float* w1p = (float*)(ws + WS_W1P);
  int stride = gridDim.x * blockDim.x;
  for (int i = blockIdx.x * blockDim.x + threadIdx.x; i < W1P_FLOATS;
       i += stride) {
    int row = i / W1P_STRIDE;       // 0 .. 60*16-1
    int k   = i - row * W1P_STRIDE;
    w1p[i] = (k < 257) ? W1[(size_t)row * 257 + k] : 0.0f;
  }
}

__global__ void k_count(const int* __restrict__ ply, int* ws, int n) {
  int i = blockIdx.x * blockDim.x + threadIdx.x;
  if (i < n) atomicAdd(&ws[WS_COUNTS + ply[i]], 1);  // BUCKET_SIZE == 1
}

__global__ void k_scan(int* ws) {
  if (threadIdx.x != 0 || blockIdx.x != 0) return;
  int soff = 0, toff = 0;
  for (int b = 0; b < NB; ++b) {
    ws[WS_TSTART + b] = toff;
    ws[WS_CURSOR + b] = soff;
    int c  = ws[WS_COUNTS + b];
    int nt = (c + 15) >> 4;         // 16-sample tiles, padded
    toff += nt;
    soff += nt * 16;
  }
  ws[WS_TSTART + NB] = toff;
}

__global__ void k_scatter(const int* __restrict__ ply, int* ws, int n) {
  int i = blockIdx.x * blockDim.x + threadIdx.x;
  if (i < n) {
    int b   = ply[i];
    int pos = atomicAdd(&ws[WS_CURSOR + b], 1);
    ws[WS_PERM + pos] = i;
  }
}

__global__ __launch_bounds__(32) void k_eval(
    const float* __restrict__ x_base, const float* __restrict__ x_pa,
    const float* __restrict__ mobility,
    const float* __restrict__ b1,
    const float* __restrict__ W2, const float* __restrict__ b2,
    const float* __restrict__ Wout, const float* __restrict__ bout,
    const int* __restrict__ ws, float* __restrict__ out) {
  __shared__ float l1s[16][17];   // raw z1 per (sample, neuron); padded stride
  __shared__ float l2s[16][65];   // screlu(z2) per (sample, neuron); padded

  const int t    = blockIdx.x;    // tile id
  const int lane = threadIdx.x;   // wave32
  const int r    = lane & 15;     // row (A: M, B: N)
  const int half = lane >> 4;
  const int kOff = half * 2;      // K sub-offset within a 4-wide K block

  // tile -> bucket (uniform scalar search over 60 entries)
  int b = 0;
  {
    const int* ts = ws + WS_TSTART;
    while (b < NB - 1 && ts[b + 1] <= t) ++b;
  }

  // Padded rows (perm == -1) alias sample 0: finite garbage in discarded
  // rows only -> no masking needed anywhere, loads stay branch-free and
  // EXEC stays all-ones for every WMMA.
  const int  s     = ws[WS_PERM + t * 16 + r];
  const bool valid = (s >= 0);
  const int  s0    = valid ? s : 0;

  const float* xb  = x_base + (size_t)s0 * 128;
  const float* xp  = x_pa   + (size_t)s0 * 128;
  const float* w1r = (const float*)(ws + WS_W1P) +
                     ((size_t)b * 16 + r) * W1P_STRIDE;  // B[k][n] = W1[b][n][k]

  // ---------------- L1: [16x260] x [260x16], dual accumulators --------------
  v8f c1a = {}, c1b = {};
  #pragma unroll 4
  for (int kk = 0; kk < 32; kk += 2) {           // K = 0..127 (x_base)
    const int kg0 = 4 * kk + kOff, kg1 = kg0 + 4;
    c1a = wmma4(*(const v2f*)(xb + kg0), *(const v2f*)(w1r + kg0), c1a);
    c1b = wmma4(*(const v2f*)(xb + kg1), *(const v2f*)(w1r + kg1), c1b);
  }
  #pragma unroll 4
  for (int kk = 0; kk < 32; kk += 2) {           // K = 128..255 (x_pa)
    const int kg0 = 4 * kk + kOff, kg1 = kg0 + 4;
    c1a = wmma4(*(const v2f*)(xp + kg0), *(const v2f*)(w1r + 128 + kg0), c1a);
    c1b = wmma4(*(const v2f*)(xp + kg1), *(const v2f*)(w1r + 128 + kg1), c1b);
  }
  {                                              // K = 256 mob; 257..259 zeros
    float mob = fminf(mobility[s0] * (7.0f / 255.0f), 1.0f);
    v2f a;
    a.x = (half == 0) ? mob : 0.0f;
    a.y = 0.0f;
    v2f w = *(const v2f*)(w1r + 256 + kOff);     // padded weights: zeros past 256
    c1a = wmma4(a, w, c1a);
  }
  v8f c1 = c1a + c1b;

  // D layout: lane L column N=L&15; VGPR j -> sample M = j + 8*(L>>4)
  const float bias1 = b1[b * 16 + r];
  #pragma unroll
  for (int j = 0; j < 8; ++j) l1s[j + 8 * half][r] = c1[j] + bias1;
  __syncthreads();

  // ---------------- L2: [16x32] x [32x64] = 8 k-steps x 4 N-tiles -----------
  v8f c2[4] = {{}, {}, {}, {}};
  const float* w2b = W2 + (size_t)b * 64 * 32;
  #pragma unroll
  for (int kk = 0; kk < 8; ++kk) {
    const int kg = 4 * kk + kOff;
    v2f a;  // A2[m][k]: k<16 -> clip(z^2), k>=16 -> clip(z); m = r
    {
      float z0 = l1s[r][kg & 15];
      float z1 = l1s[r][(kg + 1) & 15];
      a.x = (kg < 16)     ? fminf(z0 * z0, 1.0f) : fminf(fmaxf(z0, 0.0f), 1.0f);
      a.y = (kg + 1 < 16) ? fminf(z1 * z1, 1.0f) : fminf(fmaxf(z1, 0.0f), 1.0f);
    }
    #pragma unroll
    for (int nt = 0; nt < 4; ++nt) {
      v2f w = *(const v2f*)(w2b + ((size_t)(nt * 16 + r)) * 32 + kg);  // aligned
      c2[nt] = wmma4(a, w, c2[nt]);
    }
  }

  #pragma unroll
  for (int nt = 0; nt < 4; ++nt) {
    const float bias2 = b2[b * 64 + nt * 16 + r];
    #pragma unroll
    for (int j = 0; j < 8; ++j) {
      float z  = c2[nt][j] + bias2;
      float av = fminf(fmaxf(z, 0.0f), 1.0f);
      l2s[j + 8 * half][nt * 16 + r] = av * av;   // screlu
    }
  }
  __syncthreads();

  // ------------- Output layer (N=1): VALU dot of 320 feats, split by half ---
  const float* wo = Wout + (size_t)b * 320;
  float partial = 0.0f;
  if (half == 0) {  // features 0..159: l2(64) + x_base[0..95]
    #pragma unroll 8
    for (int o = 0; o < 64; ++o) partial += l2s[r][o] * wo[o];
    #pragma unroll 8
    for (int i = 0; i < 96; ++i) partial += xb[i] * wo[64 + i];
  } else {          // features 160..319: x_base[96..127] + x_pa(128)
    #pragma unroll 8
    for (int i = 96; i < 128; ++i) partial += xb[i] * wo[64 + i];
    #pragma unroll 8
    for (int i = 0; i < 128; ++i) partial += xp[i] * wo[192 + i];
  }
  float total = partial + __shfl_xor(partial, 16, 32);
  if (half == 0 && valid) out[s] = total + bout[b];
}

extern "C" void kernel_launch(void* const* d_in, const int* in_sizes, int n_in,
                              void* d_out, int out_size, void* d_ws,
                              size_t ws_size, hipStream_t stream) {
  const float* x_base   = (const float*)d_in[0];
  const float* x_pa     = (const float*)d_in[1];
  const float* mobility = (const float*)d_in[2];
  const int*   ply      = (const int*)d_in[3];
  const float* W1       = (const float*)d_in[4];
  const float* b1       = (const float*)d_in[5];
  const float* W2       = (const float*)d_in[6];
  const float* b2       = (const float*)d_in[7];
  const float* Wout     = (const float*)d_in[8];
  const float* bout     = (const float*)d_in[9];
  float* out            = (float*)d_out;
  int*   ws             = (int*)d_ws;

  const int B        = in_sizes[0] / 128;           // x_base is [B,128]
  const int maxTiles = (B + 15) / 16 + NB;          // padded tile upper bound
  const int permCap  = maxTiles * 16;
  const int thr      = 256;

  k_init<<<(permCap + thr - 1) / thr, thr, 0, stream>>>(ws, permCap);
  k_repack<<<(W1P_FLOATS + thr - 1) / thr, thr, 0, stream>>>(W1, ws);
  k_count<<<(B + thr - 1) / thr, thr, 0, stream>>>(ply, ws, B);
  k_scan<<<1, 1, 0, stream>>>(ws);
  k_scatter<<<(B + thr - 1) / thr, thr, 0, stream>>>(ply, ws, B);
  k_eval<<<maxTiles, 32, 0, stream>>>(x_base, x_pa, mobility, b1, W2, b2,
                                      Wout, bout, ws, out);
}